// DynamicFixedQuantizer_74732430950903
// MI455X (gfx1250) — compile-verified
//
#include <hip/hip_runtime.h>
#include <stdint.h>

// ---------------------------------------------------------------------------
// DynamicFixedQuantizer forward: two-pass streaming kernel for MI455X (gfx1250)
//   pass 1: global overflow/underflow counts  (read 537MB)
//   pass 2: stochastic round + clamp          (read 537MB, write 537MB, NT)
// HBM-bound: ~1.61 GB @ 23.3 TB/s => ~69 us floor. No matrix op -> no WMMA.
// ---------------------------------------------------------------------------

typedef float v4f __attribute__((ext_vector_type(4)));

#define QFWD_BITS   8
#define QR_MAX      0.01f
#define QSIGMA0     0.25f          // 2^(6-8)

// PCG-style hash -> uniform [0,1) with 24-bit mantissa, keyed by element index.
// Deterministic independent of launch geometry (harness requirement).
__device__ __forceinline__ float qz_u01(uint32_t s) {
    s = s * 747796405u + 2891336453u;
    uint32_t w = ((s >> ((s >> 28u) + 4u)) ^ s) * 277803737u;
    w = (w >> 22u) ^ w;
    return (float)(w >> 8u) * (1.0f / 16777216.0f);
}

__global__ void qz_zero_counters(uint32_t* __restrict__ cnt) {
    cnt[0] = 0u;   // overflow count
    cnt[1] = 0u;   // underflow count
}

// ---------------- Pass 1: count overflow / underflow --------------------
__global__ __launch_bounds__(256) void
qz_pass1_count(const float* __restrict__ x, uint32_t* __restrict__ cnt, int n) {
    const float t_max = QSIGMA0 * 128.0f - QSIGMA0;   // sigma*half - sigma
    const float t_min = -QSIGMA0 * 128.0f;            // -sigma*half
    const float h_max = 0.5f * t_max;
    const float h_min = 0.5f * t_min;

    uint32_t c_over = 0u, c_under = 0u;

    const int nvec   = n >> 2;
    const int stride = (int)(gridDim.x * blockDim.x);
    const int tid    = (int)(blockIdx.x * blockDim.x + threadIdx.x);
    const v4f* __restrict__ xv = (const v4f*)x;

    for (int i = tid; i < nvec; i += stride) {
        // speculative L2 prefetch of next grid-stride iteration (global_prefetch_b8)
        if (i + stride < nvec) __builtin_prefetch((const void*)(xv + i + stride), 0, 0);
        v4f v = xv[i];                                // global_load_b128
#pragma unroll
        for (int k = 0; k < 4; ++k) {
            float f = v[k];
            c_over  += (uint32_t)(f > t_max) + (uint32_t)(f < t_min);
            c_under += (uint32_t)(f > h_max) + (uint32_t)(f < h_min);
        }
    }
    for (int i = (nvec << 2) + tid; i < n; i += stride) {   // tail (unused here: n%4==0)
        float f = x[i];
        c_over  += (uint32_t)(f > t_max) + (uint32_t)(f < t_min);
        c_under += (uint32_t)(f > h_max) + (uint32_t)(f < h_min);
    }

    // wave32 shuffle reduction (MI455X is wave32 -- 5 steps, width 32)
#pragma unroll
    for (int off = 16; off > 0; off >>= 1) {
        c_over  += __shfl_down(c_over,  off, 32);
        c_under += __shfl_down(c_under, off, 32);
    }

    __shared__ uint32_t ls[2];
    if (threadIdx.x == 0) { ls[0] = 0u; ls[1] = 0u; }
    __syncthreads();
    if ((threadIdx.x & 31u) == 0u) {          // one lane per wave
        atomicAdd(&ls[0], c_over);
        atomicAdd(&ls[1], c_under);
    }
    __syncthreads();
    if (threadIdx.x == 0) {                   // one global atomic pair per block
        atomicAdd(&cnt[0], ls[0]);
        atomicAdd(&cnt[1], ls[1]);
    }
}

// ---------------- Pass 2: stochastic round + clamp ----------------------
__global__ __launch_bounds__(256) void
qz_pass2_quant(const float* __restrict__ x, float* __restrict__ out,
               const uint32_t* __restrict__ cnt, int n) {
    // Recompute sigma branchlessly from the global counts.
    // Matches reference rounding: int count -> f32 cast -> divide by f32(n).
    const float fn        = (float)n;
    const float overflow  = (float)cnt[0] / fn;
    const float underflow = (float)cnt[1] / fn;
    float sigma = QSIGMA0;
    sigma = (overflow > QR_MAX) ? (sigma * 2.0f)
          : ((underflow < QR_MAX) ? (sigma * 0.5f) : sigma);
    const float t_max     = sigma * 128.0f - sigma;
    const float t_min     = -sigma * 128.0f;
    const float inv_sigma = 1.0f / sigma;     // sigma is a power of two: exact

    const int nvec   = n >> 2;
    const int stride = (int)(gridDim.x * blockDim.x);
    // Reverse block order: pass 1 streamed forward, so the 192MB L2 holds the
    // tail of x when pass 2 starts -- harvest those hits first.
    const int rbid = (int)(gridDim.x - 1u - blockIdx.x);
    const int tid  = rbid * (int)blockDim.x + (int)threadIdx.x;

    const v4f* __restrict__ xv = (const v4f*)x;
    v4f* __restrict__ ov = (v4f*)out;

    for (int i = tid; i < nvec; i += stride) {
        if (i + stride < nvec) __builtin_prefetch((const void*)(xv + i + stride), 0, 0);
        v4f v = xv[i];
        v4f r;
        const uint32_t base = (uint32_t)i << 2;
#pragma unroll
        for (int k = 0; k < 4; ++k) {
            // fma(x, 1/sigma, u) == round(x/sigma + u) exactly (x/sigma exact),
            // identical to the reference's  floor(data/sigma + r) * sigma.
            float t = floorf(fmaf(v[k], inv_sigma, qz_u01(base + (uint32_t)k))) * sigma;
            r[k] = fminf(fmaxf(t, t_min), t_max);
        }
        __builtin_nontemporal_store(r, ov + i);   // global_store_b128 th:NT
    }
    for (int i = (nvec << 2) + tid; i < n; i += stride) {
        float t = floorf(fmaf(x[i], inv_sigma, qz_u01((uint32_t)i))) * sigma;
        out[i] = fminf(fmaxf(t, t_min), t_max);
    }
}

// ---------------------------------------------------------------------------
extern "C" void kernel_launch(void* const* d_in, const int* in_sizes, int n_in,
                              void* d_out, int out_size, void* d_ws, size_t ws_size,
                              hipStream_t stream) {
    const float* x   = (const float*)d_in[0];
    float*       out = (float*)d_out;
    uint32_t*    cnt = (uint32_t*)d_ws;       // 2 x u32 counters in workspace
    const int n = in_sizes[0];                // 32*2048*2048 = 134,217,728

    const int threads = 256;                  // 8 wave32s per workgroup
    int nvec   = n >> 2;
    int blocks = (nvec + threads - 1) / threads;
    if (blocks > 8192) blocks = 8192;         // grid-stride; ~16 float4 per thread
    if (blocks < 1)    blocks = 1;

    hipLaunchKernelGGL(qz_zero_counters, dim3(1), dim3(1), 0, stream, cnt);
    hipLaunchKernelGGL(qz_pass1_count,   dim3(blocks), dim3(threads), 0, stream, x, cnt, n);
    hipLaunchKernelGGL(qz_pass2_quant,   dim3(blocks), dim3(threads), 0, stream, x, out, cnt, n);
}